// DeformationLoss_80547816669322
// MI455X (gfx1250) — compile-verified
//
#include <hip/hip_runtime.h>
#include <math.h>

typedef __attribute__((ext_vector_type(2))) float v2f;
typedef __attribute__((ext_vector_type(4))) float v4f;
typedef __attribute__((ext_vector_type(8))) float v8f;
typedef int i32x4 __attribute__((vector_size(16)));   // matches builtin's V4i

#define GLOBAL_AS __attribute__((address_space(1)))
#define LDS_AS    __attribute__((address_space(3)))

// ---------------------------------------------------------------------------
// xor-lane butterfly add: single ds_swizzle_b32 (group-of-32 xor mode) per
// step on wave32; avoids the cndmask/bpermute sequence __shfl_xor generates.
// offset encoding: [4:0]=and_mask(0x1f), [9:5]=or_mask(0), [14:10]=xor_mask.
// ---------------------------------------------------------------------------
template <int XORMASK>
__device__ __forceinline__ float swz_xor_add(float v) {
#if __has_builtin(__builtin_amdgcn_ds_swizzle)
    int o = __builtin_amdgcn_ds_swizzle(__float_as_int(v), (XORMASK << 10) | 0x1f);
    return v + __int_as_float(o);
#else
    return v + __shfl_xor(v, XORMASK, 32);
#endif
}

// ---------------------------------------------------------------------------
// Wave-level sum of 64 f32 (2/lane) via CDNA5 V_WMMA_F32_16X16X4_F32:
// D = ones(16x4) * B(4x16) -> column sums; A=ones makes it layout-agnostic,
// so summing D row 0 across 16 lanes yields the total. EXEC must be all 1s.
// ---------------------------------------------------------------------------
__device__ __forceinline__ float wave_sum64_wmma(float a0, float a1) {
    v2f A = {1.0f, 1.0f};
    v2f B = {a0, a1};
    v8f C = {};
    v8f D = __builtin_amdgcn_wmma_f32_16x16x4_f32(
        /*neg_a=*/false, A, /*neg_b=*/false, B,
        /*c_mod=*/(short)0, C, /*reuse_a=*/false, /*reuse_b=*/false);
    float v = D[0];
    v = swz_xor_add<8>(v);
    v = swz_xor_add<4>(v);
    v = swz_xor_add<2>(v);
    v = swz_xor_add<1>(v);
    return v;   // every lane holds the wave total
}

// ---------------------------------------------------------------------------
// Async stage of one patch's cot row (E floats) into LDS.
// Uses CDNA5 GLOBAL_LOAD_ASYNC_TO_LDS_B128 + s_wait_asynccnt when available.
// ---------------------------------------------------------------------------
__device__ __forceinline__ void stage_cot_to_lds(const float* __restrict__ src,
                                                 float* s_dst, int E) {
    const int tid = threadIdx.x;
    const int bs  = blockDim.x;
#if __has_builtin(__builtin_amdgcn_global_load_async_to_lds_b128)
    const int nvec = E >> 2;  // float4 chunks (16B aligned src/dst)
    GLOBAL_AS i32x4* g = (GLOBAL_AS i32x4*)src;
    LDS_AS    i32x4* l = (LDS_AS i32x4*)s_dst;
    for (int j = tid; j < nvec; j += bs) {
        __builtin_amdgcn_global_load_async_to_lds_b128(g + j, l + j, 0, 0);
    }
    for (int j = (nvec << 2) + tid; j < E; j += bs) s_dst[j] = src[j];
#if __has_builtin(__builtin_amdgcn_s_wait_asynccnt)
    __builtin_amdgcn_s_wait_asynccnt(0);
#else
    asm volatile("s_wait_asynccnt 0x0" ::: "memory");
#endif
#else
    for (int j = tid; j < E; j += bs) s_dst[j] = src[j];
#endif
    __syncthreads();
}

// ---------------------------------------------------------------------------
// K0: init edge map to -1
// ---------------------------------------------------------------------------
__global__ void k_init_map(int* __restrict__ edgeMap, int n) {
    int i = blockIdx.x * blockDim.x + threadIdx.x;
    int stride = gridDim.x * blockDim.x;
    for (; i < n; i += stride) edgeMap[i] = -1;
}

// ---------------------------------------------------------------------------
// K1: transformed[p,n] = R[p,n] * v[p,n] + t[p,n]
// ---------------------------------------------------------------------------
__global__ void k_transform(const float* __restrict__ verts,
                            const float* __restrict__ rot,
                            const float* __restrict__ trans,
                            float* __restrict__ out, int PN) {
    int i = blockIdx.x * blockDim.x + threadIdx.x;
    if (i >= PN) return;
    const float* R = rot + i * 9;
    const float* v = verts + i * 3;
    const float* t = trans + i * 3;
    float x = v[0], y = v[1], z = v[2];
    out[i * 3 + 0] = fmaf(R[0], x, fmaf(R[1], y, fmaf(R[2], z, t[0])));
    out[i * 3 + 1] = fmaf(R[3], x, fmaf(R[4], y, fmaf(R[5], z, t[1])));
    out[i * 3 + 2] = fmaf(R[6], x, fmaf(R[7], y, fmaf(R[8], z, t[2])));
}

// ---------------------------------------------------------------------------
// K2: cotangent weights per (patch=blockIdx.y, face). cot[p][g*M+f], /4 folded.
// area = sqrt(max(s(s-A)(s-B)(s-C), 1e-12)) exactly as reference.
// ---------------------------------------------------------------------------
__global__ void k_cot(const float* __restrict__ verts,
                      const int* __restrict__ faces,
                      float* __restrict__ cot, int N, int M) {
    int f = blockIdx.x * blockDim.x + threadIdx.x;
    if (f >= M) return;
    int p = blockIdx.y;
    int i0 = faces[3 * f + 0], i1 = faces[3 * f + 1], i2 = faces[3 * f + 2];
    const float* v0 = verts + (p * N + i0) * 3;
    const float* v1 = verts + (p * N + i1) * 3;
    const float* v2 = verts + (p * N + i2) * 3;
    float ax = v1[0] - v2[0], ay = v1[1] - v2[1], az = v1[2] - v2[2];
    float bx = v0[0] - v2[0], by = v0[1] - v2[1], bz = v0[2] - v2[2];
    float cx = v0[0] - v1[0], cy = v0[1] - v1[1], cz = v0[2] - v1[2];
    float A = sqrtf(ax * ax + ay * ay + az * az);
    float B = sqrtf(bx * bx + by * by + bz * bz);
    float C = sqrtf(cx * cx + cy * cy + cz * cz);
    float s = 0.5f * (A + B + C);
    float prod = s * (s - A) * (s - B) * (s - C);
    float area = sqrtf(fmaxf(prod, 1e-12f));
    float inv = 0.25f / area;
    float A2 = A * A, B2 = B * B, C2 = C * C;
    int base = p * 3 * M;
    cot[base + 0 * M + f] = (B2 + C2 - A2) * inv;
    cot[base + 1 * M + f] = (A2 + C2 - B2) * inv;
    cot[base + 2 * M + f] = (A2 + B2 - C2) * inv;
}

// ---------------------------------------------------------------------------
// K3: "last edge index wins" scatter. Edge ids: (v0,v1)->f, (v1,v2)->M+f,
// (v0,v2)->2M+f, matching concat([faces[:,:2], faces[:,1:], faces[:,::2]]).
// atomicMax == sequential last-write (max id) semantics; order-independent.
// ---------------------------------------------------------------------------
__global__ void k_scatter(const int* __restrict__ faces,
                          int* __restrict__ edgeMap, int N, int M) {
    int f = blockIdx.x * blockDim.x + threadIdx.x;
    if (f >= M) return;
    int f0 = faces[3 * f + 0], f1 = faces[3 * f + 1], f2 = faces[3 * f + 2];
    atomicMax(&edgeMap[f0 * N + f1], f);
    atomicMax(&edgeMap[f1 * N + f2], M + f);
    atomicMax(&edgeMap[f0 * N + f2], 2 * M + f);
}

// ---------------------------------------------------------------------------
// Per-edge ARAP contribution; cot gathers come from the LDS-staged row.
// ---------------------------------------------------------------------------
__device__ __forceinline__ float edge_contrib(const float* __restrict__ s_cot,
                                              const int* __restrict__ edgeMap,
                                              int N, int a, int b,
                                              const float* Va, const float* Vb,
                                              const float* Ta, const float* Tb,
                                              const float* Ra, const float* ta) {
    int eF = edgeMap[a * N + b];
    int eB = edgeMap[b * N + a];
    float w = s_cot[eF];
    if (eB >= 0) w += s_cot[eB];
    float ex = Va[0] - Vb[0], ey = Va[1] - Vb[1], ez = Va[2] - Vb[2];
    float rex = fmaf(Ra[0], ex, fmaf(Ra[1], ey, fmaf(Ra[2], ez, ta[0])));
    float rey = fmaf(Ra[3], ex, fmaf(Ra[4], ey, fmaf(Ra[5], ez, ta[1])));
    float rez = fmaf(Ra[6], ex, fmaf(Ra[7], ey, fmaf(Ra[8], ez, ta[2])));
    float dx = rex - (Ta[0] - Tb[0]);
    float dy = rey - (Ta[1] - Tb[1]);
    float dz = rez - (Ta[2] - Tb[2]);
    return w * (dx * dx + dy * dy + dz * dz);
}

// ---------------------------------------------------------------------------
// K4: ARAP edge loss. Grid (ceil(M/BS), P); one face (3 edges) per thread.
// No integer division anywhere; shared gathers reused across the 3 edges.
// ---------------------------------------------------------------------------
__global__ void k_edge_loss(const float* __restrict__ verts,
                            const int* __restrict__ faces,
                            const float* __restrict__ rot,
                            const float* __restrict__ trans,
                            const float* __restrict__ transformed,
                            const float* __restrict__ cot,
                            const int* __restrict__ edgeMap,
                            float* __restrict__ partials,
                            int N, int M) {
    extern __shared__ float s_cot[];
    const int E = 3 * M;
    const int p = blockIdx.y;

    stage_cot_to_lds(cot + (size_t)p * E, s_cot, E);

    int f = blockIdx.x * blockDim.x + threadIdx.x;
    float acc0 = 0.0f, acc1 = 0.0f;
    if (f < M) {
        int f0 = faces[3 * f + 0], f1 = faces[3 * f + 1], f2 = faces[3 * f + 2];
        const float* V0 = verts + (p * N + f0) * 3;
        const float* V1 = verts + (p * N + f1) * 3;
        const float* V2 = verts + (p * N + f2) * 3;
        const float* T0 = transformed + (p * N + f0) * 3;
        const float* T1 = transformed + (p * N + f1) * 3;
        const float* T2 = transformed + (p * N + f2) * 3;
        const float* R0 = rot + (p * N + f0) * 9;
        const float* R1 = rot + (p * N + f1) * 9;
        const float* t0 = trans + (p * N + f0) * 3;
        const float* t1 = trans + (p * N + f1) * 3;
        // edge (f0,f1): source vertex f0
        acc0 += edge_contrib(s_cot, edgeMap, N, f0, f1, V0, V1, T0, T1, R0, t0);
        // edge (f1,f2): source vertex f1
        acc1 += edge_contrib(s_cot, edgeMap, N, f1, f2, V1, V2, T1, T2, R1, t1);
        // edge (f0,f2): source vertex f0
        acc0 += edge_contrib(s_cot, edgeMap, N, f0, f2, V0, V2, T0, T2, R0, t0);
    }
    // all lanes converged here -> WMMA wave reduction (EXEC all 1s)
    float wsum = wave_sum64_wmma(acc0, acc1);
    __shared__ float smem[8];
    int lane = threadIdx.x & 31, wid = threadIdx.x >> 5;
    if (lane == 0) smem[wid] = wsum;
    __syncthreads();
    if (threadIdx.x == 0) {
        float s = 0.0f;
        int nw = blockDim.x >> 5;
        for (int k = 0; k < nw; ++k) s += smem[k];   // fixed order: deterministic
        partials[blockIdx.y * gridDim.x + blockIdx.x] = s;
    }
}

// ---------------------------------------------------------------------------
// K5: deterministic final reduction + handle losses. Single block.
// ---------------------------------------------------------------------------
__global__ void k_finalize(const float* __restrict__ partials, int nblk,
                           const float* __restrict__ transformed,
                           const int* __restrict__ moving,
                           const int* __restrict__ stat,
                           const float* __restrict__ handle,
                           float* __restrict__ out, int P, int N, int H) {
    float a0 = 0.0f, a1 = 0.0f;
    int par = 0;
    for (int i = threadIdx.x; i < nblk; i += blockDim.x, par ^= 1) {
        if (par) a1 += partials[i]; else a0 += partials[i];
    }
    float wsum = wave_sum64_wmma(a0, a1);
    __shared__ float smem[8];
    int lane = threadIdx.x & 31, wid = threadIdx.x >> 5;
    if (lane == 0) smem[wid] = wsum;
    __syncthreads();
    if (threadIdx.x == 0) {
        float s = 0.0f;
        int nw = blockDim.x >> 5;
        for (int k = 0; k < nw; ++k) s += smem[k];
        out[0] = s / (float)P;   // mean over patches of per-patch edge sums

        float ml = 0.0f;
        for (int k = 0; k < H; ++k) {
            int pi = moving[2 * k + 0], vi = moving[2 * k + 1];
            const float* t = transformed + (pi * N + vi) * 3;
            const float* h = handle + pi * 3;
            for (int c = 0; c < 3; ++c) { float d = t[c] - h[c]; ml += d * d; }
        }
        out[1] = ml / (float)(H * 3);

        float sl = 0.0f;
        for (int k = 0; k < H; ++k) {
            int pi = stat[2 * k + 0], vi = stat[2 * k + 1];
            const float* t = transformed + (pi * N + vi) * 3;
            const float* h = handle + pi * 3;
            for (int c = 0; c < 3; ++c) { float d = t[c] - h[c]; sl += d * d; }
        }
        out[2] = sl / (float)(H * 3);
    }
}

// ---------------------------------------------------------------------------
// Host launcher
// ---------------------------------------------------------------------------
extern "C" void kernel_launch(void* const* d_in, const int* in_sizes, int n_in,
                              void* d_out, int out_size, void* d_ws, size_t ws_size,
                              hipStream_t stream) {
    const float* verts  = (const float*)d_in[0];   // (P,N,3)
    const int*   faces  = (const int*)  d_in[1];   // (M,3)
    const float* rot    = (const float*)d_in[2];   // (P,N,3,3)
    const float* trans  = (const float*)d_in[3];   // (P,N,3)
    const int*   moving = (const int*)  d_in[4];   // (H,2)
    const int*   statc  = (const int*)  d_in[5];   // (H,2)
    const float* handle = (const float*)d_in[6];   // (P,3)
    // d_in[7] = alternation (forward values identical either way)

    const int P = in_sizes[6] / 3;
    const int N = in_sizes[0] / (3 * P);
    const int M = in_sizes[1] / 3;
    const int H = in_sizes[4] / 2;
    const int E = 3 * M;

    // workspace layout
    float* w_transformed = (float*)d_ws;                      // P*N*3 floats
    float* w_cot         = w_transformed + (size_t)P * N * 3; // P*E floats
    int*   w_edgeMap     = (int*)(w_cot + (size_t)P * E);     // N*N ints
    float* w_partials    = (float*)(w_edgeMap + (size_t)N * N);

    float* out = (float*)d_out;

    const int BS = 256;
    const int FBLK = (M + BS - 1) / BS;          // face blocks per patch
    const int NPART = FBLK * P;                  // total partials

    k_init_map<<<(N * N + BS - 1) / BS, BS, 0, stream>>>(w_edgeMap, N * N);
    k_transform<<<(P * N + BS - 1) / BS, BS, 0, stream>>>(verts, rot, trans,
                                                          w_transformed, P * N);
    k_cot<<<dim3(FBLK, P), BS, 0, stream>>>(verts, faces, w_cot, N, M);
    k_scatter<<<(M + BS - 1) / BS, BS, 0, stream>>>(faces, w_edgeMap, N, M);
    k_edge_loss<<<dim3(FBLK, P), BS, (size_t)E * sizeof(float), stream>>>(
        verts, faces, rot, trans, w_transformed, w_cot, w_edgeMap,
        w_partials, N, M);
    k_finalize<<<1, BS, 0, stream>>>(w_partials, NPART, w_transformed,
                                     moving, statc, handle, out, P, N, H);
}